// TGN_41738492182812
// MI455X (gfx1250) — compile-verified
//
#include <hip/hip_runtime.h>
#include <math.h>

namespace {

constexpr int D   = 172;   // embed dim
constexpr int HD  = 86;    // head dim
constexpr int MSG = 688;   // 4*D
constexpr int D3  = 516;   // 3*D
constexpr int KN  = 20;    // neighbors
constexpr int TD  = 11;    // ceil(D/16) col tiles

typedef float v2f __attribute__((ext_vector_type(2)));
typedef float v8f __attribute__((ext_vector_type(8)));

__device__ __forceinline__ v8f wmma4(v2f a, v2f b, v8f c) {
  // V_WMMA_F32_16X16X4_F32 : D = A(16x4) * B(4x16) + C, fp32 end-to-end
  return __builtin_amdgcn_wmma_f32_16x16x4_f32(false, a, false, b, (short)0, c,
                                               false, false);
}

__device__ __forceinline__ float sigm(float x) { return 1.f / (1.f + __expf(-x)); }

#define ZERO8 {0.f,0.f,0.f,0.f,0.f,0.f,0.f,0.f}

// ---------------------------------------------------------------- pos map
__global__ void pos_init(int* __restrict__ pos, int N) {
  int i = blockIdx.x * 256 + threadIdx.x;
  if (i < N) pos[i] = -1;
}
__global__ void pos_scatter(const int* __restrict__ uniq, int* __restrict__ pos, int U) {
  int i = blockIdx.x * 256 + threadIdx.x;
  if (i < U) pos[uniq[i]] = i;
}

// ---------------------------------------------------------------- fused GRU
// grid (U/16, TD), 1 wave/block. Computes h_new[U,D].
__global__ __launch_bounds__(32) void gru_kernel(
    const float* __restrict__ msg, const float* __restrict__ memory,
    const int* __restrict__ uniq, const float* __restrict__ W_ih,
    const float* __restrict__ W_hh, const float* __restrict__ b_ih,
    const float* __restrict__ b_hh, float* __restrict__ h_new) {
  const int lane = threadIdx.x, half = lane >> 4, l16 = lane & 15;
  const int r0 = blockIdx.x * 16, c0 = blockIdx.y * 16;
  const int col = c0 + l16;
  const int colc = col < D ? col : D - 1;   // clamped for loads; stores guarded

  v8f ir = ZERO8, iz = ZERO8, in_ = ZERO8;
  v8f hr = ZERO8, hz = ZERO8, hn = ZERO8;

  // gi = msg @ W_ih.T  (B[k][n] = W_ih[n][k] -> contiguous float2 along k)
  const float* Arow = msg + (size_t)(r0 + l16) * MSG + 2 * half;
  const float* Br = W_ih + (size_t)colc * MSG + 2 * half;
  const float* Bz = W_ih + (size_t)(colc + D) * MSG + 2 * half;
  const float* Bn = W_ih + (size_t)(colc + 2 * D) * MSG + 2 * half;
  for (int k = 0; k < MSG; k += 4) {
    v2f a = *(const v2f*)(Arow + k);
    ir  = wmma4(a, *(const v2f*)(Br + k), ir);
    iz  = wmma4(a, *(const v2f*)(Bz + k), iz);
    in_ = wmma4(a, *(const v2f*)(Bn + k), in_);
  }
  // gh = memory[uniq] @ W_hh.T
  const int urow = uniq[r0 + l16];
  const float* Ah = memory + (size_t)urow * D + 2 * half;
  const float* Cr = W_hh + (size_t)colc * D + 2 * half;
  const float* Cz = W_hh + (size_t)(colc + D) * D + 2 * half;
  const float* Cn = W_hh + (size_t)(colc + 2 * D) * D + 2 * half;
  for (int k = 0; k < D; k += 4) {
    v2f a = *(const v2f*)(Ah + k);
    hr = wmma4(a, *(const v2f*)(Cr + k), hr);
    hz = wmma4(a, *(const v2f*)(Cz + k), hz);
    hn = wmma4(a, *(const v2f*)(Cn + k), hn);
  }
  const float bir = b_ih[colc], biz = b_ih[colc + D], bin = b_ih[colc + 2 * D];
  const float bhr = b_hh[colc], bhz = b_hh[colc + D], bhn = b_hh[colc + 2 * D];
  for (int e = 0; e < 8; ++e) {
    int row = r0 + e + 8 * half;
    float rg = sigm(ir[e] + bir + hr[e] + bhr);
    float zg = sigm(iz[e] + biz + hz[e] + bhz);
    float ng = tanhf(in_[e] + bin + rg * (hn[e] + bhn));
    float hold = memory[(size_t)uniq[row] * D + colc];
    if (col < D) h_new[(size_t)row * D + col] = (1.f - zg) * ng + zg * hold;
  }
}

// ---------------------------------------------------------------- node_feat gather
__global__ void gather_nodefeat(const int* __restrict__ src, const int* __restrict__ dst,
                                const int* __restrict__ neg, int Bn,
                                const float* __restrict__ memory,
                                const float* __restrict__ node_feats,
                                const float* __restrict__ h_new,
                                const int* __restrict__ pos,
                                float* __restrict__ nfeat) {
  int i = blockIdx.x * 256 + threadIdx.x;
  int total = 3 * Bn * D;
  if (i >= total) return;
  int row = i / D, c = i % D;
  int node = row < Bn ? src[row] : (row < 2 * Bn ? dst[row - Bn] : neg[row - 2 * Bn]);
  int p = pos[node];
  float mv = (p >= 0) ? h_new[(size_t)p * D + c] : memory[(size_t)node * D + c];
  nfeat[i] = mv + node_feats[(size_t)node * D + c];
}

// ---------------------------------------------------------------- qbias = cos(b) @ Wq[D:]
__global__ void qbias_kernel(const float* __restrict__ t2v_b, const float* __restrict__ Wq,
                             float* __restrict__ qbias) {
  int n = threadIdx.x;
  if (n >= D) return;
  float s = 0.f;
  for (int c = 0; c < D; ++c) s += cosf(t2v_b[c]) * Wq[(size_t)(D + c) * D + n];
  qbias[n] = s;
}

// ---------------------------------------------------------------- q = nfeat @ Wq[:D] + qbias
__global__ __launch_bounds__(32) void q_gemm(const float* __restrict__ A,
                                             const float* __restrict__ Wq,
                                             const float* __restrict__ qbias,
                                             float* __restrict__ q) {
  const int lane = threadIdx.x, half = lane >> 4, l16 = lane & 15;
  const int r0 = blockIdx.x * 16, c0 = blockIdx.y * 16;
  const int col = c0 + l16, colc = col < D ? col : D - 1;
  v8f acc = ZERO8;
  const float* Arow = A + (size_t)(r0 + l16) * D + 2 * half;
  for (int k = 0; k < D; k += 4) {
    v2f a = *(const v2f*)(Arow + k);
    v2f b;
    b.x = Wq[(size_t)(k + 2 * half) * D + colc];
    b.y = Wq[(size_t)(k + 2 * half + 1) * D + colc];
    acc = wmma4(a, b, acc);
  }
  float qb = qbias[colc];
  for (int e = 0; e < 8; ++e) {
    int row = r0 + e + 8 * half;
    if (col < D) q[(size_t)row * D + col] = acc[e] + qb;
  }
}

// ---------------------------------------------------------------- fused temporal attention
// one block per target row n; kv_in built in LDS, K/V projection via WMMA from LDS,
// then scores/softmax/ctx in LDS. Only ctx[3B,D] leaves the WGP.
__global__ __launch_bounds__(256) void attn_kernel(
    const int* __restrict__ nbrs, const int* __restrict__ nbr_eidx,
    const float* __restrict__ nbr_times, const float* __restrict__ t_arr,
    const float* __restrict__ memory, const float* __restrict__ node_feats,
    const float* __restrict__ edge_feats, const float* __restrict__ h_new,
    const int* __restrict__ pos, const float* __restrict__ t2v_w,
    const float* __restrict__ t2v_b, const float* __restrict__ Wk,
    const float* __restrict__ Wv, const float* __restrict__ qg,
    float* __restrict__ ctxg, int Bn) {
  extern __shared__ float smem[];
  float* kv   = smem;              // [32][516]  kv_in tile (rows >= KN zero)
  float* kb   = kv + 32 * D3;      // [32][172]  projected K (rows < KN valid)
  float* vb   = kb + 32 * D;       // [32][172]  projected V
  float* qrow = vb + 32 * D;       // [172]
  float* sc   = qrow + D;          // [2*KN] scores
  float* aw   = sc + 2 * KN;       // [2*KN] softmax weights
  __shared__ int   s_nbr[KN];
  __shared__ int   s_eix[KN];
  __shared__ float s_dt[KN];

  const int n = blockIdx.x, tid = threadIdx.x;
  if (tid < KN) {
    s_nbr[tid] = nbrs[(size_t)n * KN + tid];
    s_eix[tid] = nbr_eidx[(size_t)n * KN + tid];
    s_dt[tid]  = t_arr[n % Bn] - nbr_times[(size_t)n * KN + tid];
  }
  if (tid < D) qrow[tid] = qg[(size_t)n * D + tid];
  __syncthreads();

  // build kv_in = [nbr_feat | edge_feat | cos(dt*w+b)]
  for (int idx = tid; idx < 32 * D3; idx += 256) {
    int j = idx / D3, c = idx - j * D3;
    float v = 0.f;
    if (j < KN) {
      if (c < D) {
        int nb = s_nbr[j]; int p = pos[nb];
        v = ((p >= 0) ? h_new[(size_t)p * D + c] : memory[(size_t)nb * D + c]) +
            node_feats[(size_t)nb * D + c];
      } else if (c < 2 * D) {
        v = edge_feats[(size_t)s_eix[j] * D + (c - D)];
      } else {
        int cc = c - 2 * D;
        v = cosf(s_dt[j] * t2v_w[cc] + t2v_b[cc]);
      }
    }
    kv[idx] = v;
  }
  __syncthreads();

  // K/V projection: 2 M-tiles x 22 N-tiles (11 for K, 11 for V) over 8 waves
  const int wave = tid >> 5, lane = tid & 31, half = lane >> 4, l16 = lane & 15;
  for (int t = wave; t < 2 * 2 * TD; t += 8) {
    int mt = t & 1, nt = t >> 1;                 // nt 0..21
    const float* W = (nt < TD) ? Wk : Wv;
    float* outb    = (nt < TD) ? kb : vb;
    int c0 = ((nt < TD) ? nt : nt - TD) * 16;
    int col = c0 + l16, colc = col < D ? col : D - 1;
    const float* Arow = kv + (size_t)(mt * 16 + l16) * D3 + 2 * half;
    v8f acc = ZERO8;
    for (int k = 0; k < D3; k += 4) {
      v2f a = *(const v2f*)(Arow + k);            // ds_load_b64 from LDS
      v2f b;
      b.x = W[(size_t)(k + 2 * half) * D + colc];
      b.y = W[(size_t)(k + 2 * half + 1) * D + colc];
      acc = wmma4(a, b, acc);
    }
    for (int e = 0; e < 8; ++e) {
      int row = mt * 16 + e + 8 * half;
      if (row < KN && col < D) outb[row * D + col] = acc[e];
    }
  }
  __syncthreads();

  // scores = q . k / sqrt(HD), mask nbrs==0
  if (tid < 2 * KN) {
    int h = tid / KN, j = tid - h * KN;
    const float* kr = kb + j * D + h * HD;
    const float* qr = qrow + h * HD;
    float s = 0.f;
    for (int d = 0; d < HD; ++d) s += qr[d] * kr[d];
    s *= (1.0f / sqrtf((float)HD));
    if (s_nbr[j] == 0) s = -1e9f;
    sc[tid] = s;
  }
  __syncthreads();
  if (tid < 2) {                                  // softmax per head
    float mx = -1e30f;
    for (int j = 0; j < KN; ++j) mx = fmaxf(mx, sc[tid * KN + j]);
    float sum = 0.f;
    for (int j = 0; j < KN; ++j) { float e = __expf(sc[tid * KN + j] - mx); aw[tid * KN + j] = e; sum += e; }
    float inv = 1.f / sum;
    for (int j = 0; j < KN; ++j) aw[tid * KN + j] *= inv;
  }
  __syncthreads();
  for (int c = tid; c < D; c += 256) {            // ctx = attn @ v
    int h = c / HD;
    float acc = 0.f;
    for (int j = 0; j < KN; ++j) acc += aw[h * KN + j] * vb[j * D + c];
    ctxg[(size_t)n * D + c] = acc;
  }
}

// ---------------------------------------------------------------- merge layer 1 (relu([ctx|nfeat]@Wm1+bm1))
__global__ __launch_bounds__(32) void merge1(const float* __restrict__ ctx,
                                             const float* __restrict__ nfeat,
                                             const float* __restrict__ Wm1,
                                             const float* __restrict__ bm1,
                                             float* __restrict__ hbuf) {
  const int lane = threadIdx.x, half = lane >> 4, l16 = lane & 15;
  const int r0 = blockIdx.x * 16, c0 = blockIdx.y * 16;
  const int col = c0 + l16, colc = col < D ? col : D - 1;
  const size_t rowoff = (size_t)(r0 + l16) * D;
  v8f acc = ZERO8;
  for (int k = 0; k < 2 * D; k += 4) {
    int kk = k + 2 * half;                        // D%4==0 -> both halves same side
    const float* Ap = (kk < D) ? (ctx + rowoff + kk) : (nfeat + rowoff + (kk - D));
    v2f a = *(const v2f*)Ap;
    v2f b;
    b.x = Wm1[(size_t)kk * D + colc];
    b.y = Wm1[(size_t)(kk + 1) * D + colc];
    acc = wmma4(a, b, acc);
  }
  float bb = bm1[colc];
  for (int e = 0; e < 8; ++e) {
    int row = r0 + e + 8 * half;
    if (col < D) hbuf[(size_t)row * D + col] = fmaxf(acc[e] + bb, 0.f);
  }
}

// ---------------------------------------------------------------- merge layer 2 (h@Wm2+bm2)
__global__ __launch_bounds__(32) void merge2(const float* __restrict__ hbuf,
                                             const float* __restrict__ Wm2,
                                             const float* __restrict__ bm2,
                                             float* __restrict__ zbuf) {
  const int lane = threadIdx.x, half = lane >> 4, l16 = lane & 15;
  const int r0 = blockIdx.x * 16, c0 = blockIdx.y * 16;
  const int col = c0 + l16, colc = col < D ? col : D - 1;
  v8f acc = ZERO8;
  const float* Arow = hbuf + (size_t)(r0 + l16) * D + 2 * half;
  for (int k = 0; k < D; k += 4) {
    v2f a = *(const v2f*)(Arow + k);
    v2f b;
    b.x = Wm2[(size_t)(k + 2 * half) * D + colc];
    b.y = Wm2[(size_t)(k + 2 * half + 1) * D + colc];
    acc = wmma4(a, b, acc);
  }
  float bb = bm2[colc];
  for (int e = 0; e < 8; ++e) {
    int row = r0 + e + 8 * half;
    if (col < D) zbuf[(size_t)row * D + col] = acc[e] + bb;
  }
}

// ---------------------------------------------------------------- link hidden: relu(zs@Ws + zx@Wd + bs + bd)
__global__ __launch_bounds__(32) void link_hidden(const float* __restrict__ zbuf,
                                                  const float* __restrict__ Ws,
                                                  const float* __restrict__ Wd,
                                                  const float* __restrict__ bs,
                                                  const float* __restrict__ bd,
                                                  float* __restrict__ hh, int Bn) {
  const int lane = threadIdx.x, half = lane >> 4, l16 = lane & 15;
  const int r0 = blockIdx.x * 16, c0 = blockIdx.y * 16, which = blockIdx.z;
  const int col = c0 + l16, colc = col < D ? col : D - 1;
  v8f acc = ZERO8;
  const float* A1 = zbuf + (size_t)(r0 + l16) * D + 2 * half;                    // z_src
  const float* A2 = zbuf + (size_t)(which + 1) * Bn * D + (size_t)(r0 + l16) * D + 2 * half;
  for (int k = 0; k < D; k += 4) {
    v2f a = *(const v2f*)(A1 + k);
    v2f b;
    b.x = Ws[(size_t)(k + 2 * half) * D + colc];
    b.y = Ws[(size_t)(k + 2 * half + 1) * D + colc];
    acc = wmma4(a, b, acc);
  }
  for (int k = 0; k < D; k += 4) {
    v2f a = *(const v2f*)(A2 + k);
    v2f b;
    b.x = Wd[(size_t)(k + 2 * half) * D + colc];
    b.y = Wd[(size_t)(k + 2 * half + 1) * D + colc];
    acc = wmma4(a, b, acc);
  }
  float bb = bs[colc] + bd[colc];
  for (int e = 0; e < 8; ++e) {
    int row = r0 + e + 8 * half;
    if (col < D) hh[(size_t)which * Bn * D + (size_t)row * D + col] = fmaxf(acc[e] + bb, 0.f);
  }
}

// ---------------------------------------------------------------- final sigmoid GEMV
__global__ void out_kernel(const float* __restrict__ hh, const float* __restrict__ Wo,
                           const float* __restrict__ bo, float* __restrict__ out, int total) {
  int i = blockIdx.x * 256 + threadIdx.x;
  if (i >= total) return;
  const float* h = hh + (size_t)i * D;
  float s = bo[0];
  for (int c = 0; c < D; ++c) s += h[c] * Wo[c];
  out[i] = sigm(s);
}

}  // namespace

extern "C" void kernel_launch(void* const* d_in, const int* in_sizes, int n_in,
                              void* d_out, int out_size, void* d_ws, size_t ws_size,
                              hipStream_t stream) {
  const float* memory     = (const float*)d_in[0];
  const float* node_feats = (const float*)d_in[1];
  const float* edge_feats = (const float*)d_in[2];
  const float* unique_msg = (const float*)d_in[3];
  const float* t_arr      = (const float*)d_in[4];
  const float* nbr_times  = (const float*)d_in[5];
  const float* W_ih = (const float*)d_in[6];
  const float* W_hh = (const float*)d_in[7];
  const float* b_ih = (const float*)d_in[8];
  const float* b_hh = (const float*)d_in[9];
  const float* t2v_w = (const float*)d_in[10];
  const float* t2v_b = (const float*)d_in[11];
  const float* Wq  = (const float*)d_in[12];
  const float* Wk  = (const float*)d_in[13];
  const float* Wv  = (const float*)d_in[14];
  const float* Wm1 = (const float*)d_in[15];
  const float* bm1 = (const float*)d_in[16];
  const float* Wm2 = (const float*)d_in[17];
  const float* bm2 = (const float*)d_in[18];
  const float* Ws  = (const float*)d_in[19];
  const float* bs  = (const float*)d_in[20];
  const float* Wd  = (const float*)d_in[21];
  const float* bd  = (const float*)d_in[22];
  const float* Wo  = (const float*)d_in[23];
  const float* bo  = (const float*)d_in[24];
  const int* uniq = (const int*)d_in[25];
  const int* src  = (const int*)d_in[26];
  const int* dst  = (const int*)d_in[27];
  const int* neg  = (const int*)d_in[28];
  const int* nbrs = (const int*)d_in[29];
  const int* neix = (const int*)d_in[30];

  const int N  = in_sizes[0] / D;
  const int U  = in_sizes[25];
  const int Bn = in_sizes[4];
  const int M3 = 3 * Bn;

  // workspace carve-out
  char* wsp = (char*)d_ws;
  size_t off = 0;
  auto carve = [&](size_t bytes) {
    void* p = wsp + off;
    off = (off + bytes + 255) & ~(size_t)255;
    return p;
  };
  int*   pos   = (int*)  carve((size_t)N * 4);
  float* h_new = (float*)carve((size_t)U * D * 4);
  float* nfeat = (float*)carve((size_t)M3 * D * 4);
  float* qbias = (float*)carve((size_t)D * 4);
  float* qbuf  = (float*)carve((size_t)M3 * D * 4);
  float* ctx   = (float*)carve((size_t)M3 * D * 4);
  float* hbuf  = (float*)carve((size_t)M3 * D * 4);
  float* zbuf  = (float*)carve((size_t)M3 * D * 4);
  float* hh    = (float*)carve((size_t)2 * Bn * D * 4);
  (void)ws_size; (void)n_in; (void)out_size;

  pos_init<<<(N + 255) / 256, 256, 0, stream>>>(pos, N);
  pos_scatter<<<(U + 255) / 256, 256, 0, stream>>>(uniq, pos, U);

  gru_kernel<<<dim3(U / 16, TD), 32, 0, stream>>>(unique_msg, memory, uniq, W_ih,
                                                  W_hh, b_ih, b_hh, h_new);

  gather_nodefeat<<<((size_t)M3 * D + 255) / 256, 256, 0, stream>>>(
      src, dst, neg, Bn, memory, node_feats, h_new, pos, nfeat);

  qbias_kernel<<<1, 192, 0, stream>>>(t2v_b, Wq, qbias);
  q_gemm<<<dim3(M3 / 16, TD), 32, 0, stream>>>(nfeat, Wq, qbias, qbuf);

  const size_t lds_bytes = (size_t)(32 * D3 + 2 * 32 * D + D + 4 * KN) * 4;
  attn_kernel<<<M3, 256, lds_bytes, stream>>>(nbrs, neix, nbr_times, t_arr, memory,
                                              node_feats, edge_feats, h_new, pos,
                                              t2v_w, t2v_b, Wk, Wv, qbuf, ctx, Bn);

  merge1<<<dim3(M3 / 16, TD), 32, 0, stream>>>(ctx, nfeat, Wm1, bm1, hbuf);
  merge2<<<dim3(M3 / 16, TD), 32, 0, stream>>>(hbuf, Wm2, bm2, zbuf);

  link_hidden<<<dim3(Bn / 16, TD, 2), 32, 0, stream>>>(zbuf, Ws, Wd, bs, bd, hh, Bn);
  out_kernel<<<(2 * Bn + 255) / 256, 256, 0, stream>>>(hh, Wo, bo, (float*)d_out, 2 * Bn);
}